// GaussianMixtureLayer_27857157881993
// MI455X (gfx1250) — compile-verified
//
#include <hip/hip_runtime.h>
#include <math.h>

#define NS 16384   // samples
#define NK 16      // components
#define ND 128     // features
#define SDP 130    // padded diff row stride (k_maha), even => v2f-aligned
#define TSP 34     // transposed sample stride (k_outer), even => v2f-aligned
#define NPART 4    // deterministic sample-split partials for k_outer

constexpr float EPSC    = 1e-3f;
constexpr float JITTER  = 1e-6f;
constexpr float LOG_2PI = 1.8378770664093453f;

typedef float v2f __attribute__((ext_vector_type(2)));
typedef float v8f __attribute__((ext_vector_type(8)));

__device__ __forceinline__ v2f ld2(const float* p) {
    return *reinterpret_cast<const v2f*>(p);
}

// ---------------------------------------------------------------- init
__global__ void k_init(const float* __restrict__ means0, const float* __restrict__ covs0,
                       float* __restrict__ cur_means, float* __restrict__ cur_covs,
                       int* __restrict__ done) {
    int i = blockIdx.x * blockDim.x + threadIdx.x;
    for (int j = i; j < NK * ND * ND; j += gridDim.x * blockDim.x) cur_covs[j] = covs0[j];
    if (i < NK * ND) cur_means[i] = means0[i];
    if (i == 0) *done = 0;
}

__global__ void k_save_prev(const float* __restrict__ cur_means, float* __restrict__ prev_means,
                            const int* __restrict__ done) {
    if (done && *done) return;
    int i = blockIdx.x * blockDim.x + threadIdx.x;
    if (i < NK * ND) prev_means[i] = cur_means[i];
}

// ------------------------------------------- Gauss-Jordan inverse + logdet
__global__ void k_invert(const float* __restrict__ covs, float* __restrict__ inv,
                         float* __restrict__ logdet, const int* __restrict__ done) {
    if (done && *done) return;
    extern __shared__ float smem[];
    float* A = smem;              // ND*ND working copy
    float* B = smem + ND * ND;    // ND*ND inverse accumulator
    __shared__ float fac[ND];
    __shared__ float sld;
    const int k = blockIdx.x, tid = threadIdx.x;
    for (int i = tid; i < ND * ND; i += blockDim.x) {
        A[i] = covs[(size_t)k * ND * ND + i];
        B[i] = ((i >> 7) == (i & 127)) ? 1.0f : 0.0f;
    }
    if (tid == 0) sld = 0.0f;
    __syncthreads();
    for (int p = 0; p < ND; ++p) {
        float pv = A[p * ND + p];
        float rp = 1.0f / pv;
        if (tid == 0) sld += logf(fabsf(pv));
        __syncthreads();
        for (int j = tid; j < ND; j += blockDim.x) { A[p * ND + j] *= rp; B[p * ND + j] *= rp; }
        __syncthreads();
        for (int r = tid; r < ND; r += blockDim.x) fac[r] = (r == p) ? 0.0f : A[r * ND + p];
        __syncthreads();
        for (int idx = tid; idx < ND * ND; idx += blockDim.x) {
            int r = idx >> 7, j = idx & 127;
            if (r != p) {
                float f = fac[r];
                A[idx] -= f * A[p * ND + j];
                B[idx] -= f * B[p * ND + j];
            }
        }
        __syncthreads();
    }
    for (int i = tid; i < ND * ND; i += blockDim.x) inv[(size_t)k * ND * ND + i] = B[i];
    if (tid == 0) logdet[k] = sld;
}

// ------------------------------------- w[n,k] = diff^T inv_k diff  (WMMA f32)
// inv stored pair-interleaved in LDS so each B fragment is one ds_load_b64;
// diff rows padded to SDP to kill the 16-way A-load bank conflict.
__global__ void __launch_bounds__(128)
k_maha(const float* __restrict__ x, const float* __restrict__ means,
       const float* __restrict__ inv, float* __restrict__ w_out,
       const int* __restrict__ done) {
    if (done && *done) return;
    extern __shared__ float smem[];
    float* sinvP = smem;            // ND*ND, layout [(row/2)*ND + col][row&1]
    float* sdiff = smem + ND * ND;  // 64 * SDP
    const int k   = blockIdx.y;
    const int n0  = blockIdx.x * 64;
    const int tid = threadIdx.x;
    for (int i = tid; i < ND * ND; i += 128) {
        int row = i >> 7, col = i & 127;
        sinvP[(((row >> 1) * ND) + col) * 2 + (row & 1)] = inv[(size_t)k * ND * ND + i];
    }
    for (int i = tid; i < 64 * ND; i += 128) {
        int s = i >> 7, f = i & 127;
        sdiff[s * SDP + f] = x[(size_t)(n0 + s) * ND + f] - means[k * ND + f];
    }
    __syncthreads();
    const int lane = tid & 31, wave = tid >> 5;
    const int half = lane >> 4, l16 = lane & 15;
    const int srow = wave * 16 + l16;      // A-frag row (sample)
    v8f acc[8] = {};
    for (int kk = 0; kk < ND; kk += 4) {
        const int kb = kk + half * 2;          // lanes 16-31 hold K=2,3
        const int kbHalf = (kk >> 1) + half;   // kb/2
        v2f a = ld2(&sdiff[srow * SDP + kb]);  // contiguous pair, 8B aligned
#pragma unroll
        for (int j = 0; j < 8; ++j) {
            int col = j * 16 + l16;
            v2f b = ld2(&sinvP[(kbHalf * ND + col) * 2]);  // one b64 per fragment
            acc[j] = __builtin_amdgcn_wmma_f32_16x16x4_f32(false, a, false, b,
                                                           (short)0, acc[j], false, false);
        }
    }
    // w[m] = sum_{j,v} acc[j][v] * diff[m][col];  C layout: M = v + 8*half, N = l16
    float part[8];
#pragma unroll
    for (int v = 0; v < 8; ++v) part[v] = 0.0f;
#pragma unroll
    for (int j = 0; j < 8; ++j)
#pragma unroll
        for (int v = 0; v < 8; ++v) {
            int m = v + half * 8;
            part[v] += acc[j][v] * sdiff[(wave * 16 + m) * SDP + j * 16 + l16];
        }
#pragma unroll
    for (int v = 0; v < 8; ++v) {
        float s = part[v];
        s += __shfl_xor(s, 1);
        s += __shfl_xor(s, 2);
        s += __shfl_xor(s, 4);
        s += __shfl_xor(s, 8);
        part[v] = s;
    }
    if (l16 == 0) {
#pragma unroll
        for (int v = 0; v < 8; ++v) {
            int m = v + half * 8;
            w_out[(size_t)(n0 + wave * 16 + m) * NK + k] = part[v];
        }
    }
}

// --------------------------------------------------- reductions over samples
__global__ void k_reduce_sw(const float* __restrict__ w, float* __restrict__ sw,
                            const int* __restrict__ done) {
    if (done && *done) return;
    __shared__ float red[256];
    int k = blockIdx.x, tid = threadIdx.x;
    float s = 0.0f;
    for (int n = tid; n < NS; n += 256) s += w[(size_t)n * NK + k];
    red[tid] = s; __syncthreads();
    for (int off = 128; off > 0; off >>= 1) { if (tid < off) red[tid] += red[tid + off]; __syncthreads(); }
    if (tid == 0) sw[k] = red[0];
}

__global__ void k_reduce_swx(const float* __restrict__ w, const float* __restrict__ x,
                             float* __restrict__ swx, const int* __restrict__ done) {
    if (done && *done) return;
    __shared__ float red[256];
    int k = blockIdx.x >> 7, d = blockIdx.x & 127, tid = threadIdx.x;
    float s = 0.0f;
    for (int n = tid; n < NS; n += 256) s += w[(size_t)n * NK + k] * x[(size_t)n * ND + d];
    red[tid] = s; __syncthreads();
    for (int off = 128; off > 0; off >>= 1) { if (tid < off) red[tid] += red[tid + off]; __syncthreads(); }
    if (tid == 0) swx[k * ND + d] = red[0];
}

// --------------------------- S2_k = Diff_k^T diag(w) Diff_k  (WMMA f32, heavy)
// Transposed LDS staging (feature-major, sample stride TSP) so A and B frags
// are single v2f LDS loads; w premultiplied at staging. Samples split into
// NPART deterministic partials for occupancy. Grid: (2 rowHalf, NK*NPART).
__global__ void __launch_bounds__(128)
k_outer(const float* __restrict__ x, const float* __restrict__ means,
        const float* __restrict__ w, float* __restrict__ S2p,
        const int* __restrict__ done) {
    if (done && *done) return;
    extern __shared__ float smem[];
    float* sdT = smem;             // ND * TSP  : diff^T
    float* swT = smem + ND * TSP;  // ND * TSP  : (w*diff)^T
    __shared__ float swv[32];
    const int rh   = blockIdx.x;
    const int k    = blockIdx.y >> 2;        // NPART == 4
    const int part = blockIdx.y & 3;
    const int tid  = threadIdx.x;
    const int lane = tid & 31, wave = tid >> 5;
    const int half = lane >> 4, l16 = lane & 15;
    const int rowBase = rh * 64 + wave * 16;
    const int nStart = part * (NS / NPART);
    const int nEnd   = nStart + (NS / NPART);
    v8f acc[8] = {};
    for (int n0 = nStart; n0 < nEnd; n0 += 32) {
        if (n0 + 32 < nEnd)   // speculative prefetch of next chunk (one line/thread)
            __builtin_prefetch(&x[(size_t)(n0 + 32) * ND + tid * 32], 0, 3);
        __syncthreads();
        if (tid < 32) swv[tid] = w[(size_t)(n0 + tid) * NK + k];
        __syncthreads();
        for (int i = tid; i < 32 * ND; i += 128) {
            int s = i >> 7, f = i & 127;
            float d = x[(size_t)(n0 + s) * ND + f] - means[k * ND + f];
            sdT[f * TSP + s] = d;
            swT[f * TSP + s] = swv[s] * d;
        }
        __syncthreads();
        for (int c = 0; c < 32; c += 4) {
            const int s0 = c + half * 2;                      // lanes 16-31: K=2,3
            v2f a = ld2(&sdT[(rowBase + l16) * TSP + s0]);    // A = Diff^T fragment
#pragma unroll
            for (int j = 0; j < 8; ++j) {
                int col = j * 16 + l16;
                v2f b = ld2(&swT[col * TSP + s0]);            // B = (w*Diff)^T fragment
                acc[j] = __builtin_amdgcn_wmma_f32_16x16x4_f32(false, a, false, b,
                                                               (short)0, acc[j], false, false);
            }
        }
    }
#pragma unroll
    for (int j = 0; j < 8; ++j)
#pragma unroll
        for (int v = 0; v < 8; ++v) {
            int row = rowBase + v + half * 8;
            int col = j * 16 + l16;
            S2p[(size_t)part * NK * ND * ND + (size_t)k * ND * ND + (size_t)row * ND + col]
                = acc[j][v];
        }
}

// ---------------------------------------------------------------- finalize
__global__ void k_finalize_covs(const float* __restrict__ S2p, const float* __restrict__ sw,
                                float* __restrict__ covs_out, const int* __restrict__ done) {
    if (done && *done) return;
    int i = blockIdx.x * blockDim.x + threadIdx.x;
    if (i >= NK * ND * ND) return;
    const size_t P = (size_t)NK * ND * ND;
    int k = i / (ND * ND);
    int rc = i % (ND * ND);
    float v = (S2p[i] + S2p[P + i] + S2p[2 * P + i] + S2p[3 * P + i]) / sw[k];
    if ((rc >> 7) == (rc & 127)) v += JITTER;
    covs_out[i] = v;
}

__global__ void k_finalize_means(const float* __restrict__ swx, const float* __restrict__ sw,
                                 const float* __restrict__ prev, float* __restrict__ means_out,
                                 int* __restrict__ done) {
    if (done && *done) return;
    __shared__ float red[256];
    int tid = threadIdx.x;
    float d2 = 0.0f;
    for (int i = tid; i < NK * ND; i += 256) {
        float nm = swx[i] / sw[i >> 7];
        float df = nm - prev[i];
        d2 += df * df;
        means_out[i] = nm;
    }
    red[tid] = d2; __syncthreads();
    for (int off = 128; off > 0; off >>= 1) { if (tid < off) red[tid] += red[tid + off]; __syncthreads(); }
    if (tid == 0 && done) { if (red[0] <= EPSC * EPSC) *done = 1; }
}

// loss = 0.5 * sum_k weights_k * ( N*(D*log2pi + logdet_k) + sum_n maha_nk )
__global__ void k_loss(const float* __restrict__ weights, const float* __restrict__ logdet,
                       const float* __restrict__ mahasum, float* __restrict__ out) {
    __shared__ float red[32];
    int tid = threadIdx.x;
    float t = 0.0f;
    if (tid < NK)
        t = weights[tid] * ((float)NS * ((float)ND * LOG_2PI + logdet[tid]) + mahasum[tid]);
    red[tid] = t; __syncthreads();
    for (int off = 16; off > 0; off >>= 1) { if (tid < off) red[tid] += red[tid + off]; __syncthreads(); }
    if (tid == 0) out[0] = 0.5f * red[0];
}

// ================================================================ launch
extern "C" void kernel_launch(void* const* d_in, const int* in_sizes, int n_in,
                              void* d_out, int out_size, void* d_ws, size_t ws_size,
                              hipStream_t stream) {
    const float* x       = (const float*)d_in[0];
    const float* means0  = (const float*)d_in[1];
    const float* covs0   = (const float*)d_in[2];
    const float* weights = (const float*)d_in[3];
    float* out = (float*)d_out;
    float* ws  = (float*)d_ws;

    // workspace layout (floats): ~7.0 MB total
    float* cur_means  = ws;                        // 2048
    float* prev_means = cur_means + 2048;          // 2048
    float* cur_covs   = prev_means + 2048;         // 262144
    float* inv        = cur_covs + 262144;         // 262144
    float* S2p        = inv + 262144;              // NPART * 262144
    float* wbuf       = S2p + NPART * 262144;      // NS*NK = 262144
    float* swx        = wbuf + 262144;             // 2048
    float* sw         = swx + 2048;                // 16
    float* logdet     = sw + 16;                   // 16
    int*   done       = (int*)(logdet + 16);       // 1

    float* m_fin = out + 1;
    float* c_fin = out + 1 + NK * ND;

    const size_t lds_inv  = (size_t)2 * ND * ND * sizeof(float);              // 128 KB
    const size_t lds_maha = (size_t)(ND * ND + 64 * SDP) * sizeof(float);     // ~98 KB
    const size_t lds_s2   = (size_t)(2 * ND * TSP) * sizeof(float);           // ~34 KB

    const dim3 gOuter(2, NK * NPART);

    k_init<<<dim3(1024), dim3(256), 0, stream>>>(means0, covs0, cur_means, cur_covs, done);

    for (int it = 0; it < 8; ++it) {
        k_save_prev<<<dim3(8), dim3(256), 0, stream>>>(cur_means, prev_means, done);
        k_invert<<<dim3(NK), dim3(256), lds_inv, stream>>>(cur_covs, inv, logdet, done);
        k_maha<<<dim3(NS / 64, NK), dim3(128), lds_maha, stream>>>(x, cur_means, inv, wbuf, done);
        k_reduce_sw<<<dim3(NK), dim3(256), 0, stream>>>(wbuf, sw, done);
        k_reduce_swx<<<dim3(NK * ND), dim3(256), 0, stream>>>(wbuf, x, swx, done);
        k_outer<<<gOuter, dim3(128), lds_s2, stream>>>(x, cur_means, wbuf, S2p, done);
        k_finalize_covs<<<dim3(1024), dim3(256), 0, stream>>>(S2p, sw, cur_covs, done);
        k_finalize_means<<<dim3(1), dim3(256), 0, stream>>>(swx, sw, prev_means, cur_means, done);
    }

    // final (ungated) update -> m_fin / c_fin in d_out
    k_invert<<<dim3(NK), dim3(256), lds_inv, stream>>>(cur_covs, inv, logdet, nullptr);
    k_maha<<<dim3(NS / 64, NK), dim3(128), lds_maha, stream>>>(x, cur_means, inv, wbuf, nullptr);
    k_reduce_sw<<<dim3(NK), dim3(256), 0, stream>>>(wbuf, sw, nullptr);
    k_reduce_swx<<<dim3(NK * ND), dim3(256), 0, stream>>>(wbuf, x, swx, nullptr);
    k_outer<<<gOuter, dim3(128), lds_s2, stream>>>(x, cur_means, wbuf, S2p, nullptr);
    k_finalize_covs<<<dim3(1024), dim3(256), 0, stream>>>(S2p, sw, c_fin, nullptr);
    k_finalize_means<<<dim3(1), dim3(256), 0, stream>>>(swx, sw, prev_means, m_fin, nullptr);

    // log-prob: inv(c_fin) gives both logdet (pivot products) and mahalanobis
    k_invert<<<dim3(NK), dim3(256), lds_inv, stream>>>(c_fin, inv, logdet, nullptr);
    k_maha<<<dim3(NS / 64, NK), dim3(128), lds_maha, stream>>>(x, m_fin, inv, wbuf, nullptr);
    k_reduce_sw<<<dim3(NK), dim3(256), 0, stream>>>(wbuf, sw, nullptr);   // sw = sum_n maha
    k_loss<<<dim3(1), dim3(32), 0, stream>>>(weights, logdet, sw, out);
}